// KNNRegressor_89137751261339
// MI455X (gfx1250) — compile-verified
//
#include <hip/hip_runtime.h>

typedef __attribute__((ext_vector_type(2))) float v2f;
typedef __attribute__((ext_vector_type(8))) float v8f;

#define DIMS 128
#define KTOP 32
#define NSLICE 8
#define SBSTRIDE 132   // 16 rows x 132 floats: conflict-free, 16B-aligned
#define INF __builtin_inff()

// ---------------------------------------------------------------- t^2 kernel
__global__ void tsq_kernel(const float* __restrict__ X, float* __restrict__ tsq, int N) {
  int r = blockIdx.x * blockDim.x + threadIdx.x;
  if (r >= N) return;
  const float4* row = (const float4*)(X + (size_t)r * DIMS);
  float s = 0.f;
#pragma unroll
  for (int i = 0; i < DIMS / 4; ++i) {
    float4 v = row[i];
    s += v.x * v.x + v.y * v.y + v.z * v.z + v.w * v.w;
  }
  tsq[r] = s;
}

// ------------------------------------------------- partial top-k per N-slice
__global__ void __launch_bounds__(256) knn_partial_kernel(
    const float* __restrict__ Qm, const float* __restrict__ X,
    const float* __restrict__ tsq, float* __restrict__ pd, int* __restrict__ pi,
    int N, int sliceLen) {
  __shared__ float sB[16 * SBSTRIDE];
  __shared__ float sQsq[8][16];

  const int lane = threadIdx.x & 31;
  const int w    = threadIdx.x >> 5;
  const int kh   = lane >> 4;      // which K-half of the wave
  const int nlo  = lane & 15;      // column / row-in-tile index
  const int qtile  = blockIdx.x * 128 + w * 16;
  const int slice  = blockIdx.y;
  const int nStart = slice * sliceLen;
  const int nEnd   = min(nStart + sliceLen, N);

  // ---- preload A fragments: 16 queries x 128 dims, fp32 WMMA layout
  // lane l holds Q[m=l%16][kc*4 + 2*kh], Q[m][kc*4 + 2*kh + 1] in a[kc]
  v2f a[32];
  {
    const float* qrow = Qm + (size_t)(qtile + nlo) * DIMS;
#pragma unroll
    for (int kc = 0; kc < 32; ++kc) {
      const float2 t = *(const float2*)(qrow + kc * 4 + 2 * kh);
      v2f av; av.x = t.x; av.y = t.y;
      a[kc] = av;
    }
  }

  // ---- per-query |q|^2 (two half-lanes hold complementary K halves)
  float part = 0.f;
#pragma unroll
  for (int kc = 0; kc < 32; ++kc) part += a[kc].x * a[kc].x + a[kc].y * a[kc].y;
  part += __shfl_xor(part, 16, 32);
  if (lane < 16) sQsq[w][lane] = part;
  float qs0[8];
#pragma unroll
  for (int j = 0; j < 8; ++j) qs0[j] = sQsq[w][j + 8 * kh];

  // ---- register-resident top-32: lane = slot
  float topd[16], thr[16];
  int   topi[16];
#pragma unroll
  for (int m = 0; m < 16; ++m) { topd[m] = INF; thr[m] = INF; topi[m] = 0; }

  const int nTiles = (nEnd - nStart + 15) >> 4;
  for (int t = 0; t < nTiles; ++t) {
    const int n0 = nStart + t * 16;

    __syncthreads();
    {   // cooperative stage of B tile: 16 train rows x 128 floats
      const int idx  = threadIdx.x * 8;   // 256 threads x 8 floats = 2048
      const int p    = idx >> 7;
      const int dcol = idx & 127;
      int gr = n0 + p; if (gr >= N) gr = N - 1;   // clamped rows masked via tsq
      const float4* src = (const float4*)(X + (size_t)gr * DIMS + dcol);
      float4 v0 = src[0], v1 = src[1];
      float* dst = &sB[p * SBSTRIDE + dcol];
      *(float4*)(dst)     = v0;
      *(float4*)(dst + 4) = v1;
    }
    __syncthreads();

    // ---- 16x16 cross products via native fp32 WMMA, K = 128 = 32 x 4
    v8f acc = {0.f, 0.f, 0.f, 0.f, 0.f, 0.f, 0.f, 0.f};
#pragma unroll
    for (int kc = 0; kc < 32; ++kc) {
      const float2 bt = *(const float2*)&sB[nlo * SBSTRIDE + kc * 4 + 2 * kh];
      v2f b; b.x = bt.x; b.y = bt.y;
      acc = __builtin_amdgcn_wmma_f32_16x16x4_f32(
          false, a[kc], false, b, (short)0, acc, false, false);
    }

    // ---- distances: lane holds (n = nlo, m = j + 8*kh)
    float tn = INF;
    const int nc = n0 + nlo;
    if (nc < nEnd) tn = tsq[nc];
    float dist[8];
#pragma unroll
    for (int j = 0; j < 8; ++j) dist[j] = qs0[j] + tn - 2.0f * acc[j];

    // ---- top-k update, threshold-filtered (rarely inserts)
#pragma unroll
    for (int m = 0; m < 16; ++m) {
      const float dm = __shfl(dist[m & 7], nlo + ((m >> 3) << 4), 32);
      const bool pass = (lane < 16) && (dm < thr[m]);
      unsigned msk = (unsigned)(__ballot(pass) & 0xffffffffull);
      while (msk) {
        const int c = __builtin_ctz(msk); msk &= msk - 1;
        const float dc = __shfl(dm, c, 32);
        float mx = topd[m];
#pragma unroll
        for (int o = 16; o >= 1; o >>= 1) mx = fmaxf(mx, __shfl_xor(mx, o, 32));
        if (dc < mx) {   // wave-uniform
          const unsigned mm = (unsigned)(__ballot(topd[m] == mx) & 0xffffffffull);
          const int tgt = __builtin_ctz(mm);
          if (lane == tgt) { topd[m] = dc; topi[m] = n0 + c; }
          float m2 = topd[m];
#pragma unroll
          for (int o = 16; o >= 1; o >>= 1) m2 = fmaxf(m2, __shfl_xor(m2, o, 32));
          thr[m] = m2;
        }
      }
    }
  }

  // ---- write partial top-32 per (query, slice)
#pragma unroll
  for (int m = 0; m < 16; ++m) {
    const size_t off = ((size_t)(qtile + m) * NSLICE + slice) * KTOP + lane;
    pd[off] = topd[m];
    pi[off] = topi[m];
  }
}

// ------------------------------------------- merge NSLICE partial lists
__global__ void __launch_bounds__(256) knn_merge_kernel(
    const float* __restrict__ pd, const int* __restrict__ pi,
    const float* __restrict__ y, float* __restrict__ out, int Qn) {
  const int lane = threadIdx.x & 31;
  const int q = blockIdx.x * 8 + (threadIdx.x >> 5);
  if (q >= Qn) return;

  float d[NSLICE]; int ix[NSLICE];
#pragma unroll
  for (int s2 = 0; s2 < NSLICE; ++s2) {
    const size_t off = ((size_t)q * NSLICE + s2) * KTOP + lane;
    d[s2]  = pd[off];
    ix[s2] = pi[off];
  }

  float sum = 0.f;
  for (int it = 0; it < KTOP; ++it) {
    float lmin = d[0]; int ls = 0;
#pragma unroll
    for (int s2 = 1; s2 < NSLICE; ++s2) if (d[s2] < lmin) { lmin = d[s2]; ls = s2; }
    float g = lmin;
#pragma unroll
    for (int o = 16; o >= 1; o >>= 1) g = fminf(g, __shfl_xor(g, o, 32));
    const unsigned mm = (unsigned)(__ballot(lmin == g) & 0xffffffffull);
    const int tgt = __builtin_ctz(mm);
    if (lane == tgt) {
      int sel = ix[0];
#pragma unroll
      for (int s2 = 1; s2 < NSLICE; ++s2) sel = (s2 == ls) ? ix[s2] : sel;
      sum += y[sel];
#pragma unroll
      for (int s2 = 0; s2 < NSLICE; ++s2) if (s2 == ls) d[s2] = INF;
    }
  }
#pragma unroll
  for (int o = 16; o >= 1; o >>= 1) sum += __shfl_xor(sum, o, 32);
  if (lane == 0) out[q] = sum * (1.0f / 32.0f);
}

// --------------------------------------------------------------- launcher
extern "C" void kernel_launch(void* const* d_in, const int* in_sizes, int n_in,
                              void* d_out, int out_size, void* d_ws, size_t ws_size,
                              hipStream_t stream) {
  const float* Qm = (const float*)d_in[0];
  const float* X  = (const float*)d_in[1];
  const float* y  = (const float*)d_in[2];
  const int Qn = in_sizes[0] / DIMS;   // 4096
  const int N  = in_sizes[1] / DIMS;   // 100000

  float* tsq = (float*)d_ws;
  size_t off = ((size_t)N * sizeof(float) + 255) & ~(size_t)255;
  float* pd = (float*)((char*)d_ws + off);
  size_t pdBytes = (size_t)Qn * NSLICE * KTOP * sizeof(float);
  int* pi = (int*)((char*)d_ws + off + pdBytes);
  float* out = (float*)d_out;

  tsq_kernel<<<(N + 255) / 256, 256, 0, stream>>>(X, tsq, N);

  const int sliceLen = (N + NSLICE - 1) / NSLICE;
  dim3 grid(Qn / 128, NSLICE);
  knn_partial_kernel<<<grid, 256, 0, stream>>>(Qm, X, tsq, pd, pi, N, sliceLen);

  knn_merge_kernel<<<(Qn + 7) / 8, 256, 0, stream>>>(pd, pi, y, out, Qn);
}